// neighborSearchModule_47021301957204
// MI455X (gfx1250) — compile-verified
//
#include <hip/hip_runtime.h>
#include <math.h>

#define N_PART    262144
#define GRID2     256
#define NUM_CELLS (GRID2 * GRID2)
#define KSLOT     32
#define MAXN      64

typedef __attribute__((ext_vector_type(2))) float v2f;
typedef __attribute__((ext_vector_type(8))) float v8f;

// ---- workspace layout (32-bit words) ----
// stats[0]=hMax bits, stats[1]=qMinX bits, stats[2]=qMinY bits (non-negative floats: uint order == float order)
#define WS_STATS 0
#define WS_CNT   4                                    // NUM_CELLS+1 counters
#define WS_TABLE (WS_CNT + NUM_CELLS + 1)             // (NUM_CELLS+1)*KSLOT particle indices
#define WS_LIN   (WS_TABLE + (NUM_CELLS + 1) * KSLOT) // N_PART linear cell ids
#define WS_OFL   (WS_LIN + N_PART)                    // N_PART overflow flags

__global__ void k_init(unsigned* ws) {
  long long i = (long long)blockIdx.x * blockDim.x + threadIdx.x;
  const long long tableLen = (long long)(NUM_CELLS + 1) * KSLOT;
  if (i < tableLen)       ((int*)ws)[WS_TABLE + i] = -1;
  if (i < NUM_CELLS + 1)  ws[WS_CNT + i] = 0u;
  if (i < N_PART)         ((int*)ws)[WS_OFL + i] = 0;
  if (i == 0) { ws[0] = 0u; ws[1] = 0x7f800000u; ws[2] = 0x7f800000u; }
}

// hMax = max(supports); qMin = min(positions) (subtract hMax at use site)
__global__ void k_reduce(const float* __restrict__ pos, const float* __restrict__ sup,
                         unsigned* __restrict__ ws) {
  int i = blockIdx.x * blockDim.x + threadIdx.x;
  float h = sup[i], x = pos[2 * i], y = pos[2 * i + 1];
  for (int d = 16; d; d >>= 1) {
    h = fmaxf(h, __shfl_xor(h, d, 32));
    x = fminf(x, __shfl_xor(x, d, 32));
    y = fminf(y, __shfl_xor(y, d, 32));
  }
  if ((threadIdx.x & 31) == 0) {
    atomicMax(&ws[0], __float_as_uint(h));   // non-negative floats: uint compare is exact
    atomicMin(&ws[1], __float_as_uint(x));
    atomicMin(&ws[2], __float_as_uint(y));
  }
}

__global__ void k_bin(const float* __restrict__ pos, unsigned* __restrict__ ws) {
  int i = blockIdx.x * blockDim.x + threadIdx.x;
  float h  = __uint_as_float(ws[0]);
  float qx = __uint_as_float(ws[1]) - h;
  float qy = __uint_as_float(ws[2]) - h;
  int cx = (int)ceilf((pos[2 * i]     - qx) / h); cx = min(max(cx, 0), GRID2 - 1);
  int cy = (int)ceilf((pos[2 * i + 1] - qy) / h); cy = min(max(cy, 0), GRID2 - 1);
  int c = cx + GRID2 * cy;
  ((int*)ws)[WS_LIN + i] = c;
  unsigned slot = atomicAdd(&ws[WS_CNT + c], 1u);
  if (slot < KSLOT) {
    ((int*)ws)[WS_TABLE + (long long)c * KSLOT + slot] = i;
  } else {  // mirrors reference overflow scatter into row NUM_CELLS slot 0 (race ok, jax is too)
    ((int*)ws)[WS_TABLE + (long long)NUM_CELLS * KSLOT] = i;
    ((int*)ws)[WS_OFL + i] = 1;
  }
}

// reproduce reference stable argsort: slots within a cell in ascending particle index
__global__ void k_sort(unsigned* ws) {
  int c = blockIdx.x * blockDim.x + threadIdx.x;
  if (c >= NUM_CELLS) return;
  int m = min((int)ws[WS_CNT + c], KSLOT);
  if (m <= 1) return;
  int* row = (int*)ws + WS_TABLE + (long long)c * KSLOT;
  int v[KSLOT];
  for (int k = 0; k < m; k++) v[k] = row[k];
  for (int a = 1; a < m; a++) {
    int key = v[a], b = a - 1;
    while (b >= 0 && v[b] > key) { v[b + 1] = v[b]; b--; }
    v[b + 1] = key;
  }
  for (int k = 0; k < m; k++) row[k] = v[k];
}

struct WaveShm {
  float qf[32][4];      // query features (x, y, |p|^2, 1), cell-local coords
  float qh[32];
  int   qid[32];
  float cf[288][4];     // candidate features (-2x, -2y, 1, |p|^2)
  int   ci[288];
  float dt[256];        // one 16x16 d^2 tile (row = query, 64B-aligned rows)
};

// one wave32 per cell: shared candidate set -> WMMA f32 16x16x4 distance tiles
__launch_bounds__(256)
__global__ void k_main(const float* __restrict__ pos, const float* __restrict__ sup,
                       const unsigned* __restrict__ ws, int* __restrict__ neigh,
                       int* __restrict__ cntOut, float* __restrict__ rad) {
  __shared__ WaveShm shm[8];
  const int lane = threadIdx.x & 31;
  const int wv   = threadIdx.x >> 5;
  const int c    = blockIdx.x * 8 + wv;
  WaveShm& S = shm[wv];
  const int* table = (const int*)ws + WS_TABLE;

  int mq = min((int)ws[WS_CNT + c], KSLOT);
  if (mq == 0) return;                       // wave-uniform; no block barriers used

  float h     = __uint_as_float(ws[0]);
  float refx  = (__uint_as_float(ws[1]) - h) + (float)(c & (GRID2 - 1)) * h;  // cell-local origin:
  float refy  = (__uint_as_float(ws[2]) - h) + (float)(c >> 8) * h;           // keeps |p|~2h, no cancellation

  // queries
  int mi = table[(long long)c * KSLOT + lane];
  float qx = 0.f, qy = 0.f, qhh = 0.f;
  if (lane < mq && mi >= 0) { qx = pos[2 * mi] - refx; qy = pos[2 * mi + 1] - refy; qhh = sup[mi]; }
  else mi = -1;
  S.qf[lane][0] = qx; S.qf[lane][1] = qy; S.qf[lane][2] = qx * qx + qy * qy; S.qf[lane][3] = 1.0f;
  S.qh[lane] = qhh; S.qid[lane] = mi;

  // order-preserving compaction of present candidates (reference meshgrid 'ij' cell order)
  const int offs[9] = {-257, -1, 255, -256, 0, 256, -255, 1, 257};
  int nc = 0;
  for (int k = 0; k < 9; k++) {
    int cc = min(max(c + offs[k], 0), NUM_CELLS);
    int cand = table[(long long)cc * KSLOT + lane];
    unsigned mask = (unsigned)__ballot(cand >= 0);
    if (cand >= 0) {
      int at = nc + __popc(mask & ((1u << lane) - 1u));
      float x = pos[2 * cand] - refx, y = pos[2 * cand + 1] - refy;
      S.cf[at][0] = -2.f * x; S.cf[at][1] = -2.f * y; S.cf[at][2] = 1.f; S.cf[at][3] = x * x + y * y;
      S.ci[at] = cand;
    }
    nc += __popc(mask);
  }
  int ncPad = (nc + 15) & ~15;
  if (nc + lane < ncPad) { S.cf[nc + lane][0] = 0.f; S.cf[nc + lane][1] = 0.f; S.cf[nc + lane][2] = 0.f; S.cf[nc + lane][3] = 0.f; }
  __asm__ volatile("s_wait_dscnt 0" ::: "memory");   // intra-wave LDS visibility

  const int nMT = (mq + 15) >> 4;
  const int nNT = ncPad >> 4;
  for (int mt = 0; mt < nMT; mt++) {
    // A 16x4 f32 fragment: v0 = K0(l<16)/K2(l>=16), v1 = K1/K3, M = lane&15
    int arow = mt * 16 + (lane & 15);
    v2f A;
    if (lane < 16) { A.x = S.qf[arow][0]; A.y = S.qf[arow][1]; }
    else           { A.x = S.qf[arow][2]; A.y = S.qf[arow][3]; }

    int   q    = mt * 16 + lane;                      // lanes 0..15 own one query each
    int   pid  = (lane < 16 && q < mq) ? S.qid[q] : -1;
    float myh  = (lane < 16 && q < mq) ? S.qh[q] : 1.f;
    float h2   = myh * myh;
    float rinv = 1.0f / myh;                          // one precise div per query, not per neighbor
    int cnt = 0, wr = 0;

    for (int nt = 0; nt < nNT; nt++) {
      // B 4x16 f32 fragment: rows striped across lanes, N = lane&15
      int col = nt * 16 + (lane & 15);
      v2f B;
      if (lane < 16) { B.x = S.cf[col][0]; B.y = S.cf[col][1]; }
      else           { B.x = S.cf[col][2]; B.y = S.cf[col][3]; }
      v8f Cacc = {0.f, 0.f, 0.f, 0.f, 0.f, 0.f, 0.f, 0.f};
      v8f D = __builtin_amdgcn_wmma_f32_16x16x4_f32(false, A, false, B, (short)0, Cacc, false, false);

      // D layout: VGPR r holds M=r (lanes 0-15) / M=8+r (lanes 16-31), N = lane&15
      int drow0 = (lane < 16) ? 0 : 8;
      int dcol  = lane & 15;
#pragma unroll
      for (int r = 0; r < 8; r++) S.dt[(drow0 + r) * 16 + dcol] = D[r];
      __asm__ volatile("s_wait_dscnt 0" ::: "memory");

      if (pid >= 0) {
        int lim = min(16, nc - nt * 16);
        const float4* rowp = (const float4*)&S.dt[lane * 16];   // 64B-aligned strip
#pragma unroll
        for (int j = 0; j < 4; j++) {
          float4 v4 = rowp[j];                                  // ds_load_b128
#pragma unroll
          for (int e = 0; e < 4; e++) {
            int n = j * 4 + e;
            float d2 = (e == 0) ? v4.x : (e == 1) ? v4.y : (e == 2) ? v4.z : v4.w;
            if (n < lim && d2 <= h2) {
              cnt++;
              if (wr < MAXN) {
                neigh[(long long)pid * MAXN + wr] = S.ci[nt * 16 + n];
                rad[(long long)pid * MAXN + wr]   = __builtin_amdgcn_sqrtf(fmaxf(d2, 0.f)) * rinv;
                wr++;
              }
            }
          }
        }
      }
      __asm__ volatile("s_wait_dscnt 0" ::: "memory");  // WAR before next tile store
    }
    if (pid >= 0) {
      cntOut[pid] = cnt;
      for (int k = wr; k < MAXN; k++) {
        neigh[(long long)pid * MAXN + k] = -1;
        rad[(long long)pid * MAXN + k]   = 0.f;
      }
    }
  }
}

// scalar path for overflowed particles (rank>=K) so every output row is written
__global__ void k_fallback(const float* __restrict__ pos, const float* __restrict__ sup,
                           const unsigned* __restrict__ ws, int* __restrict__ neigh,
                           int* __restrict__ cntOut, float* __restrict__ rad) {
  int i = blockIdx.x * blockDim.x + threadIdx.x;
  if (i >= N_PART) return;
  if (((const int*)ws)[WS_OFL + i] == 0) return;
  int c = ((const int*)ws)[WS_LIN + i];
  const int* table = (const int*)ws + WS_TABLE;
  float px = pos[2 * i], py = pos[2 * i + 1], h = sup[i];
  float rinv = 1.0f / h;
  const int offs[9] = {-257, -1, 255, -256, 0, 256, -255, 1, 257};
  int cnt = 0, wr = 0;
  for (int k = 0; k < 9; k++) {
    int cc = min(max(c + offs[k], 0), NUM_CELLS);
    for (int s = 0; s < KSLOT; s++) {
      int cand = table[(long long)cc * KSLOT + s];
      if (cand < 0) continue;
      float dx = pos[2 * cand] - px, dy = pos[2 * cand + 1] - py;
      float d = __builtin_amdgcn_sqrtf(dx * dx + dy * dy);
      if (d <= h) {
        cnt++;
        if (wr < MAXN) { neigh[(long long)i * MAXN + wr] = cand; rad[(long long)i * MAXN + wr] = d * rinv; wr++; }
      }
    }
  }
  cntOut[i] = cnt;
  for (int k = wr; k < MAXN; k++) { neigh[(long long)i * MAXN + k] = -1; rad[(long long)i * MAXN + k] = 0.f; }
}

extern "C" void kernel_launch(void* const* d_in, const int* in_sizes, int n_in,
                              void* d_out, int out_size, void* d_ws, size_t ws_size,
                              hipStream_t stream) {
  const float* pos = (const float*)d_in[0];
  const float* sup = (const float*)d_in[1];
  unsigned* ws = (unsigned*)d_ws;

  int*   neigh  = (int*)d_out;                                      // N x 64 int32
  int*   cntOut = (int*)d_out + (long long)N_PART * MAXN;           // N int32
  float* rad    = (float*)d_out + (long long)N_PART * MAXN + N_PART;// N x 64 f32

  long long initN = (long long)(NUM_CELLS + 1) * KSLOT;
  k_init    <<<(unsigned)((initN + 255) / 256), 256, 0, stream>>>(ws);
  k_reduce  <<<N_PART / 256, 256, 0, stream>>>(pos, sup, ws);
  k_bin     <<<N_PART / 256, 256, 0, stream>>>(pos, ws);
  k_sort    <<<(NUM_CELLS + 255) / 256, 256, 0, stream>>>(ws);
  k_main    <<<NUM_CELLS / 8, 256, 0, stream>>>(pos, sup, ws, neigh, cntOut, rad);
  k_fallback<<<N_PART / 256, 256, 0, stream>>>(pos, sup, ws, neigh, cntOut, rad);
}